// YoloLoss_42863773614661
// MI455X (gfx1250) — compile-verified
//
#include <hip/hip_runtime.h>
#include <math.h>

// ---------------- problem constants (from reference) ----------------
#define SS      13
#define NB      2
#define NC      20
#define NSAMP   4096
#define COORD_W 5.0f
#define NOOBJ_W 0.5f
#define CH      30                       // C + 5*B channels per cell
#define CELLS   (NSAMP * SS * SS)        // 692224
#define TILE    128                      // cells per tile
#define TILE_F  (TILE * CH)              // 3840 floats per tensor per tile
#define NTILES  (CELLS / TILE)           // 5408 (exact)
#define NBLK    1352                     // 5408 / 4 -> each block does 4 tiles
#define TPB     128                      // 4 wave32

// ---------------- CDNA5 async global->LDS path (guarded) ----------------
#define AS_GLOBAL __attribute__((address_space(1)))
#define AS_LDS    __attribute__((address_space(3)))

typedef int   v2i_t __attribute__((ext_vector_type(2)));  // b64 payload type
typedef float v2f_t __attribute__((ext_vector_type(2)));  // clang vector (not HIP class)

#if defined(__has_builtin)
#if __has_builtin(__builtin_amdgcn_global_load_async_to_lds_b64) && \
    __has_builtin(__builtin_amdgcn_s_wait_asynccnt)
#define USE_ASYNC 1
#endif
#endif
#ifndef USE_ASYNC
#define USE_ASYNC 0
#endif

#if USE_ASYNC
// src: global AS(1) v2i* (non-const per builtin signature), dst: LDS AS(3) v2i*
#define ASYNC_B64(gsrc, ldst)                                               \
    __builtin_amdgcn_global_load_async_to_lds_b64(                          \
        (AS_GLOBAL v2i_t*)(AS_GLOBAL void*)(void*)(gsrc),                   \
        (AS_LDS v2i_t*)(AS_LDS void*)(void*)(ldst), 0, 0)
#endif

__global__ __launch_bounds__(TPB) void yolo_stage1(
    const float* __restrict__ pred,   // CELLS * CH floats
    const float* __restrict__ targ,   // CELLS * CH floats
    float* __restrict__ part)         // NBLK partials
{
    __shared__ float sp[2][TILE_F];
    __shared__ float st[2][TILE_F];
    __shared__ float wsum[TPB / 32];

    const int  tid  = threadIdx.x;
    const float invS = 1.0f / (float)SS;

    // Stage one tile (both tensors) into LDS buffer `buf`.
    auto issue_tile = [&](int buf, int tile) {
        v2f_t* gp = (v2f_t*)(pred + (size_t)tile * TILE_F);
        v2f_t* gt = (v2f_t*)(targ + (size_t)tile * TILE_F);
        v2f_t* lp = (v2f_t*)sp[buf];
        v2f_t* lt = (v2f_t*)st[buf];
#if USE_ASYNC
#pragma unroll
        for (int i = 0; i < 15; ++i) {            // 15*128 b64 = 3840 floats
            int o = i * TPB + tid;                // contiguous b64 per wave
            ASYNC_B64(gp + o, lp + o);
            ASYNC_B64(gt + o, lt + o);
        }
#else
#pragma unroll
        for (int i = 0; i < 15; ++i) {
            int o = i * TPB + tid;
            lp[o] = __builtin_nontemporal_load(gp + o);
            lt[o] = __builtin_nontemporal_load(gt + o);
        }
#endif
    };

    float acc = 0.0f;
    int t   = blockIdx.x;
    int buf = 0;
    if (t < NTILES) issue_tile(0, t);

    while (t < NTILES) {
        const int  nxt     = t + gridDim.x;
        const bool hasnext = (nxt < NTILES);
        if (hasnext) issue_tile(buf ^ 1, nxt);    // overlap next tile's DMA
#if USE_ASYNC
        if (hasnext) __builtin_amdgcn_s_wait_asynccnt(30);  // tile t's 30 ops done
        else         __builtin_amdgcn_s_wait_asynccnt(0);
#else
        if (hasnext) {
            const int n2 = nxt + gridDim.x;       // prefetch tile-after-next
            if (n2 < NTILES) {
                __builtin_prefetch(pred + (size_t)n2 * TILE_F + tid * 16, 0, 0);
                __builtin_prefetch(targ + (size_t)n2 * TILE_F + tid * 16, 0, 0);
            }
        }
#endif
        __syncthreads();                          // all waves' tile t visible

        // ---- per-cell compute from LDS (stride-30 dwords: bank-conflict free) ----
        v2f_t pv[15], tv[15];
        {
            const v2f_t* p2 = (const v2f_t*)(sp[buf] + tid * CH);
            const v2f_t* t2 = (const v2f_t*)(st[buf] + tid * CH);
#pragma unroll
            for (int j = 0; j < 15; ++j) { pv[j] = p2[j]; tv[j] = t2[j]; }
        }
        const float* pc = (const float*)pv;
        const float* tc = (const float*)tv;

        const bool obj = tc[NC] > 0.0f;

        // target box 0 -> xyxy (matches reference rounding)
        const float tcx = tc[21] * invS, tcy = tc[22] * invS;
        const float thw = 0.5f * tc[23], thh = 0.5f * tc[24];
        const float tx1 = tcx - thw, ty1 = tcy - thh;
        const float tx2 = tcx + thw, ty2 = tcy + thh;
        const float areat = (tx2 - tx1) * (ty2 - ty1);

        auto iou_of = [&](int base) -> float {
            float cx = pc[base + 1] * invS, cy = pc[base + 2] * invS;
            float hw = 0.5f * pc[base + 3], hh = 0.5f * pc[base + 4];
            float x1 = cx - hw, y1 = cy - hh, x2 = cx + hw, y2 = cy + hh;
            float iw = fmaxf(fminf(x2, tx2) - fmaxf(x1, tx1), 0.0f);
            float ih = fmaxf(fminf(y2, ty2) - fmaxf(y1, ty1), 0.0f);
            float inter = iw * ih;
            float areap = (x2 - x1) * (y2 - y1);
            return inter / (areap + areat - inter);   // no epsilon, like reference
        };
        const float iou0 = iou_of(NC);
        const float iou1 = iou_of(NC + 5);
        const int   bs   = (iou1 > iou0) ? 1 : 0;     // jnp.argmax tie->0, NaN->0

        if (obj) {
            const int pb = NC + 5 * bs;
            float dx = pc[pb + 1] - tc[pb + 1];
            float dy = pc[pb + 2] - tc[pb + 2];
            float dw = sqrtf(pc[pb + 3]) - sqrtf(tc[pb + 3]);
            float dh = sqrtf(pc[pb + 4]) - sqrtf(tc[pb + 4]);
            float dc = pc[pb] - tc[pb];
            float cls = 0.0f;
#pragma unroll
            for (int c = 0; c < NC; ++c) {
                float d = pc[c] - tc[c];
                cls = fmaf(d, d, cls);
            }
            acc += COORD_W * (dx * dx + dy * dy + dw * dw + dh * dh) + dc * dc + cls;
        } else {
            float d0 = pc[NC] - tc[NC];
            float d1 = pc[NC + 5] - tc[NC + 5];
            acc += NOOBJ_W * (d0 * d0 + d1 * d1);
        }

        __syncthreads();                          // done reading buf before reuse
        buf ^= 1;
        t = nxt;
    }

    // ---- deterministic block reduction: wave shfl tree + LDS ----
#pragma unroll
    for (int off = 16; off > 0; off >>= 1) acc += __shfl_down(acc, off, 32);
    if ((tid & 31) == 0) wsum[tid >> 5] = acc;
    __syncthreads();
    if (tid == 0) {
        float s = 0.0f;
#pragma unroll
        for (int w = 0; w < TPB / 32; ++w) s += wsum[w];
        part[blockIdx.x] = s;
    }
}

__global__ __launch_bounds__(256) void yolo_stage2(
    const float* __restrict__ part, int n, float* __restrict__ out)
{
    __shared__ float wsum[8];
    float acc = 0.0f;
    for (int i = threadIdx.x; i < n; i += 256) acc += part[i];  // fixed order
#pragma unroll
    for (int off = 16; off > 0; off >>= 1) acc += __shfl_down(acc, off, 32);
    if ((threadIdx.x & 31) == 0) wsum[threadIdx.x >> 5] = acc;
    __syncthreads();
    if (threadIdx.x == 0) {
        float s = 0.0f;
#pragma unroll
        for (int w = 0; w < 8; ++w) s += wsum[w];
        out[0] = s * (1.0f / (float)NSAMP);
    }
}

extern "C" void kernel_launch(void* const* d_in, const int* in_sizes, int n_in,
                              void* d_out, int out_size, void* d_ws, size_t ws_size,
                              hipStream_t stream) {
    (void)in_sizes; (void)n_in; (void)out_size; (void)ws_size;
    const float* pred = (const float*)d_in[0];   // (N, S*S*(C+5B)) f32
    const float* targ = (const float*)d_in[1];   // (N, S, S, C+5B) f32
    float* part = (float*)d_ws;                  // NBLK * 4 bytes scratch

    yolo_stage1<<<NBLK, TPB, 0, stream>>>(pred, targ, part);
    yolo_stage2<<<1, 256, 0, stream>>>(part, NBLK, (float*)d_out);
}